// CombinatorialClassifier_33294586478786
// MI455X (gfx1250) — compile-verified
//
#include <hip/hip_runtime.h>
#include <math.h>

// Problem constants (from reference): B,P,K,C,D = 64,16,1000,50000,2048
#define B_ 64
#define P_ 16
#define K_ 1000
#define C_ 50000
#define D_ 2048
#define N_ (P_ * K_)   // 16000
#define ESP 1e-20f

typedef float v2f __attribute__((ext_vector_type(2)));
typedef float v8f __attribute__((ext_vector_type(8)));

// ---------------------------------------------------------------------------
// Kernel 1: partial logits = x[B,D] @ W[N,D]^T over one K-half.
// Block = 128 threads (4 waves). Each wave owns a 64(M) x 32(N) tile:
// 8 f32 16x16 accumulators (2 N sub-tiles sharing the 4 A-fragments).
// grid.y = 2 selects the K-half and the partial-logit buffer, doubling wave
// parallelism (1000 waves) without re-reading W (131 MB read exactly once).
// Per lane we load exactly the 8B K-pair its fragment slot needs
// (lanes 0-15: {k,k+1}, lanes 16-31: {k+2,k+3}) -> no selects, no over-fetch.
// ---------------------------------------------------------------------------
__global__ __launch_bounds__(128)
void gemm_wmma_f32(const float* __restrict__ X, const float* __restrict__ W,
                   float* __restrict__ logits_parts)
{
  const int tid  = threadIdx.x;
  const int wave = tid >> 5;
  const int lane = tid & 31;
  const int half = lane >> 4;   // fragment K-pair selector
  const int r    = lane & 15;   // row within 16-wide tile (A: M, B: N)

  const int n0   = blockIdx.x * 128 + wave * 32;  // this wave's N tile base
  const int koff = blockIdx.y * (D_ / 2) + 2 * half;

  const float* w0 = W + (size_t)(n0 + r)      * D_ + koff;  // B rows, N tile 0
  const float* w1 = W + (size_t)(n0 + 16 + r) * D_ + koff;  // B rows, N tile 1
  const float* xa = X + (size_t)(r)       * D_ + koff;      // A rows, M tiles
  const float* xb = X + (size_t)(16 + r)  * D_ + koff;
  const float* xc = X + (size_t)(32 + r)  * D_ + koff;
  const float* xd = X + (size_t)(48 + r)  * D_ + koff;

  v8f a00 = {}, a10 = {}, a20 = {}, a30 = {};   // N tile 0, M tiles 0..3
  v8f a01 = {}, a11 = {}, a21 = {}, a31 = {};   // N tile 1, M tiles 0..3

  #pragma unroll 2
  for (int k = 0; k < D_ / 2; k += 4) {
    v2f b0 = *(const v2f*)(w0 + k);
    v2f b1 = *(const v2f*)(w1 + k);
    v2f f0 = *(const v2f*)(xa + k);
    v2f f1 = *(const v2f*)(xb + k);
    v2f f2 = *(const v2f*)(xc + k);
    v2f f3 = *(const v2f*)(xd + k);

    a00 = __builtin_amdgcn_wmma_f32_16x16x4_f32(false, f0, false, b0, (short)0, a00, false, false);
    a10 = __builtin_amdgcn_wmma_f32_16x16x4_f32(false, f1, false, b0, (short)0, a10, false, false);
    a20 = __builtin_amdgcn_wmma_f32_16x16x4_f32(false, f2, false, b0, (short)0, a20, false, false);
    a30 = __builtin_amdgcn_wmma_f32_16x16x4_f32(false, f3, false, b0, (short)0, a30, false, false);
    a01 = __builtin_amdgcn_wmma_f32_16x16x4_f32(false, f0, false, b1, (short)0, a01, false, false);
    a11 = __builtin_amdgcn_wmma_f32_16x16x4_f32(false, f1, false, b1, (short)0, a11, false, false);
    a21 = __builtin_amdgcn_wmma_f32_16x16x4_f32(false, f2, false, b1, (short)0, a21, false, false);
    a31 = __builtin_amdgcn_wmma_f32_16x16x4_f32(false, f3, false, b1, (short)0, a31, false, false);
  }

  // C/D layout: lanes 0-15 -> N=lane, M=vgpr(0..7); lanes 16-31 -> N=lane-16,
  // M=8+vgpr. Column per lane == r, row = v + 8*half (+ 16*Mtile).
  float* dst = logits_parts + (size_t)blockIdx.y * ((size_t)B_ * N_);
  const int nA = n0 + r;
  const int nB = n0 + 16 + r;
  #pragma unroll
  for (int v = 0; v < 8; ++v) {
    const int m = v + 8 * half;
    dst[(size_t)(m)      * N_ + nA] = a00[v];
    dst[(size_t)(m + 16) * N_ + nA] = a10[v];
    dst[(size_t)(m + 32) * N_ + nA] = a20[v];
    dst[(size_t)(m + 48) * N_ + nA] = a30[v];
    dst[(size_t)(m)      * N_ + nB] = a01[v];
    dst[(size_t)(m + 16) * N_ + nB] = a11[v];
    dst[(size_t)(m + 32) * N_ + nB] = a21[v];
    dst[(size_t)(m + 48) * N_ + nB] = a31[v];
  }
}

// ---------------------------------------------------------------------------
// Kernel 2: softmax over K=1000 per (b,p); merges the two K-half partials and
// the bias, writes probs in place over the first partial buffer.
// ---------------------------------------------------------------------------
__global__ __launch_bounds__(256)
void softmax_k(const float* __restrict__ l0, const float* __restrict__ l1,
               const float* __restrict__ bias, float* __restrict__ probs)
{
  const int p = blockIdx.x;
  const int b = blockIdx.y;
  const size_t off = (size_t)b * N_ + (size_t)p * K_;

  __shared__ float red[256];
  const int t = threadIdx.x;

  float v[4];
  float mx = -3.402823466e38f;
  #pragma unroll
  for (int j = 0; j < 4; ++j) {
    const int i = t + j * 256;
    v[j] = (i < K_) ? (l0[off + i] + l1[off + i] + bias[p * K_ + i])
                    : -3.402823466e38f;
    mx = fmaxf(mx, v[j]);
  }
  red[t] = mx;
  __syncthreads();
  for (int s = 128; s > 0; s >>= 1) {
    if (t < s) red[t] = fmaxf(red[t], red[t + s]);
    __syncthreads();
  }
  mx = red[0];
  __syncthreads();

  float e[4];
  float sum = 0.f;
  #pragma unroll
  for (int j = 0; j < 4; ++j) {
    const int i = t + j * 256;
    e[j] = (i < K_) ? expf(v[j] - mx) : 0.f;
    sum += e[j];
  }
  red[t] = sum;
  __syncthreads();
  for (int s = 128; s > 0; s >>= 1) {
    if (t < s) red[t] += red[t + s];
    __syncthreads();
  }
  const float inv = 1.0f / red[0];

  #pragma unroll
  for (int j = 0; j < 4; ++j) {
    const int i = t + j * 256;
    if (i < K_) probs[off + i] = e[j] * inv;
  }
}

// ---------------------------------------------------------------------------
// Kernel 3: out[b,c] = sum_p probs[b, part[p,c]]  (probs & part L2-resident)
// ---------------------------------------------------------------------------
__global__ __launch_bounds__(256)
void gather_sum(const float* __restrict__ probs, const int* __restrict__ part,
                float* __restrict__ out)
{
  const int c = blockIdx.x * 256 + threadIdx.x;
  const int b = blockIdx.y;
  if (c >= C_) return;

  const float* prow = probs + (size_t)b * N_;
  float s = 0.f;
  #pragma unroll
  for (int p = 0; p < P_; ++p) {
    const int idx = part[(size_t)p * C_ + c];   // already offset by p*K
    s += prow[idx];
  }
  out[(size_t)b * C_ + c] = s;
}

// ---------------------------------------------------------------------------
// Kernel 4: deterministic per-row sum (no float atomics -> bitwise-stable).
// ---------------------------------------------------------------------------
__global__ __launch_bounds__(256)
void row_reduce(const float* __restrict__ out, float* __restrict__ rowsum)
{
  const int b = blockIdx.x;
  __shared__ float red[256];
  float s = 0.f;
  for (int c = threadIdx.x; c < C_; c += 256)
    s += out[(size_t)b * C_ + c];
  red[threadIdx.x] = s;
  __syncthreads();
  for (int k = 128; k > 0; k >>= 1) {
    if (threadIdx.x < k) red[threadIdx.x] += red[threadIdx.x + k];
    __syncthreads();
  }
  if (threadIdx.x == 0) rowsum[b] = red[0];
}

// ---------------------------------------------------------------------------
// Kernel 5: out = log(out / rowsum[b] + 1e-20)
// ---------------------------------------------------------------------------
__global__ __launch_bounds__(256)
void finalize(float* __restrict__ out, const float* __restrict__ rowsum)
{
  const int c = blockIdx.x * 256 + threadIdx.x;
  const int b = blockIdx.y;
  if (c >= C_) return;
  const float inv = 1.0f / rowsum[b];
  const size_t i = (size_t)b * C_ + c;
  out[i] = logf(out[i] * inv + ESP);
}

extern "C" void kernel_launch(void* const* d_in, const int* in_sizes, int n_in,
                              void* d_out, int out_size, void* d_ws, size_t ws_size,
                              hipStream_t stream)
{
  const float* X    = (const float*)d_in[0];   // (B, D)
  const float* W    = (const float*)d_in[1];   // (P*K, D)
  const float* bias = (const float*)d_in[2];   // (P*K,)
  const int*   part = (const int*)  d_in[3];   // (P, C)
  float* out = (float*)d_out;                  // (B, C)

  float* logits0 = (float*)d_ws;                          // K-half 0 partials
  float* logits1 = logits0 + (size_t)B_ * N_;             // K-half 1 partials
  float* rowsum  = logits1 + (size_t)B_ * N_;             // B floats

  // 1) f32-WMMA GEMM: grid (125 N-blocks x 2 K-halves), 4 waves/block,
  //    each wave a 64x32 tile -> 1000 waves, W streamed once from HBM.
  gemm_wmma_f32<<<dim3(N_ / 128, 2), dim3(128), 0, stream>>>(X, W, logits0);

  // 2) Merge K-half partials + bias, softmax per (b,p); probs in logits0.
  softmax_k<<<dim3(P_, B_), dim3(256), 0, stream>>>(logits0, logits1, bias, logits0);

  // 3) Gather + sum over P.
  gather_sum<<<dim3((C_ + 255) / 256, B_), dim3(256), 0, stream>>>(logits0, part, out);

  // 4) Deterministic row sums.
  row_reduce<<<dim3(B_), dim3(256), 0, stream>>>(out, rowsum);

  // 5) Normalize + log.
  finalize<<<dim3((C_ + 255) / 256, B_), dim3(256), 0, stream>>>(out, rowsum);
}